// GEN_16183436771651
// MI455X (gfx1250) — compile-verified
//
#include <hip/hip_runtime.h>
#include <limits.h>

// ---------------------------------------------------------------------------
// GENConv GNN on gfx1250 (MI455X).
// - GEMMs via v_wmma_f32_16x16x4_f32 (full f32 precision; GEMM FLOPs tiny vs
//   the 256MB/layer edge stream).
// - `e` (256MB > 192MB L2) is streamed with non-temporal hints so the hot node
//   arrays (h / mkey / denom / num / z, ~75MB) stay L2-resident for gathers
//   and the softmax-agg atomics.
// - Edge passes use 128-bit loads, 4 features per thread.
// ---------------------------------------------------------------------------

typedef __attribute__((ext_vector_type(2))) float v2f;
typedef __attribute__((ext_vector_type(4))) float v4f;
typedef __attribute__((ext_vector_type(8))) float v8f;
typedef __attribute__((ext_vector_type(4))) int v4i;

#define EPS_MSG 1e-7f
#define BN_EPS  1e-5f
#define N_NODES 50000
#define N_EDGES 1000000
#define HDIM    64
#define H2DIM   128
#define NGRAPHS 1024

__device__ __forceinline__ v8f wmma4(v2f a, v2f b, v8f c) {
  // D = A(16x4) * B(4x16) + C(16x16), all f32
  return __builtin_amdgcn_wmma_f32_16x16x4_f32(false, a, false, b, (short)0, c,
                                               false, false);
}

__device__ __forceinline__ v8f splat8(float v) {
  v8f r = {v, v, v, v, v, v, v, v};
  return r;
}

// Monotone int key for f32 (order-preserving), involution.
__device__ __forceinline__ int f32key(float f) {
  int i = __float_as_int(f);
  return i ^ ((i >> 31) & 0x7FFFFFFF);
}
__device__ __forceinline__ float f32unkey(int k) {
  return __int_as_float(k ^ ((k >> 31) & 0x7FFFFFFF));
}

// ---------------------------------------------------------------------------
// Encoder: OUT = relu( relu(X@W1 + b1) @ W2 + b2 ),  X:[M,K1], hidden/out = 64.
// One wave per 16-row tile. Intermediate staged in wave-private LDS (stride 68
// floats -> conflict-free A-layout reads). NTIO=true streams X/OUT with
// non-temporal hints (bond encoder: edge-sized arrays bypass L2 retention).
// ---------------------------------------------------------------------------
template <int K1S, bool NTIO>  // K1S = K1/4
__global__ void encoder_kernel(const float* __restrict__ X, int M, int K1,
                               const float* __restrict__ W1,
                               const float* __restrict__ B1,
                               const float* __restrict__ W2,
                               const float* __restrict__ B2,
                               float* __restrict__ OUT) {
  __shared__ float lds[4 * 16 * 68];
  const int wave = threadIdx.x >> 5;
  const int lane = threadIdx.x & 31;
  const int ln = lane & 15;
  const int hi = lane >> 4;
  const int kb = hi * 2;
  const int tile = blockIdx.x * 4 + wave;
  const int row0 = tile * 16;
  float* wlds = lds + wave * 16 * 68;

  if (row0 < M) {
    const int arow = (row0 + ln < M) ? (row0 + ln) : (M - 1);
    // A operand for GEMM1: lane holds A[row][k0+kb], A[row][k0+kb+1]
    v2f a1[K1S];
#pragma unroll
    for (int ks = 0; ks < K1S; ++ks) {
      const int kk = ks * 4 + kb;
      const float* p = X + (long long)arow * K1 + kk;
      if (NTIO) {
        a1[ks].x = __builtin_nontemporal_load(p);
        a1[ks].y = __builtin_nontemporal_load(p + 1);
      } else {
        a1[ks].x = p[0];
        a1[ks].y = p[1];
      }
    }
#pragma unroll
    for (int nt = 0; nt < 4; ++nt) {
      const int col = nt * 16 + ln;
      v8f acc = splat8(B1[col]);
#pragma unroll
      for (int ks = 0; ks < K1S; ++ks) {
        const int kk = ks * 4 + kb;
        v2f b;
        b.x = W1[kk * HDIM + col];
        b.y = W1[(kk + 1) * HDIM + col];
        acc = wmma4(a1[ks], b, acc);
      }
#pragma unroll
      for (int r = 0; r < 8; ++r) {
        const int rr = r + hi * 8;
        wlds[rr * 68 + col] = fmaxf(acc[r], 0.0f);
      }
    }
  }
  __syncthreads();
  if (row0 >= M) return;

  // GEMM2: K=64 from LDS
  v2f a2[16];
#pragma unroll
  for (int ks = 0; ks < 16; ++ks) {
    const int kk = ks * 4 + kb;
    a2[ks].x = wlds[ln * 68 + kk];
    a2[ks].y = wlds[ln * 68 + kk + 1];
  }
#pragma unroll
  for (int nt = 0; nt < 4; ++nt) {
    const int col = nt * 16 + ln;
    v8f acc = splat8(B2[col]);
#pragma unroll
    for (int ks = 0; ks < 16; ++ks) {
      const int kk = ks * 4 + kb;
      v2f b;
      b.x = W2[kk * HDIM + col];
      b.y = W2[(kk + 1) * HDIM + col];
      acc = wmma4(a2[ks], b, acc);
    }
#pragma unroll
    for (int r = 0; r < 8; ++r) {
      const int grow = row0 + r + hi * 8;
      if (grow < M) {
        float* p = OUT + (long long)grow * HDIM + col;
        const float v = fmaxf(acc[r], 0.0f);
        if (NTIO)
          __builtin_nontemporal_store(v, p);
        else
          *p = v;
      }
    }
  }
}

// ---------------------------------------------------------------------------
// Edge pass 1: msg = relu(h[src]+e)+eps ; atomic-max of msg*t into mkey[dst].
// 4 features / thread, 128-bit NT load of the e-stream.
// ---------------------------------------------------------------------------
__global__ void msg_max_kernel(const float* __restrict__ H,
                               const float* __restrict__ E_,
                               const int* __restrict__ src,
                               const int* __restrict__ dst,
                               const float* __restrict__ T, int layer,
                               int* __restrict__ mkey) {
  const long long tid = (long long)blockIdx.x * blockDim.x + threadIdx.x;
  if (tid >= (long long)N_EDGES * 16) return;
  const int e = (int)(tid >> 4);
  const int f4 = (int)(tid & 15) * 4;
  const int s = src[e], d = dst[e];
  const float t = T[layer];
  const v4f ev =
      __builtin_nontemporal_load((const v4f*)(E_ + (long long)e * HDIM + f4));
  const v4f hv = *(const v4f*)(H + (long long)s * HDIM + f4);
  int* mk = mkey + (long long)d * HDIM + f4;
#pragma unroll
  for (int j = 0; j < 4; ++j) {
    const float msg = fmaxf(hv[j] + ev[j], 0.0f) + EPS_MSG;
    atomicMax(&mk[j], f32key(msg * t));
  }
}

// ---------------------------------------------------------------------------
// Edge pass 2: ex = exp(msg*t - m[dst]); denom += ex; num += msg*ex.
// ---------------------------------------------------------------------------
__global__ void msg_sum_kernel(const float* __restrict__ H,
                               const float* __restrict__ E_,
                               const int* __restrict__ src,
                               const int* __restrict__ dst,
                               const float* __restrict__ T, int layer,
                               const int* __restrict__ mkey,
                               float* __restrict__ denom,
                               float* __restrict__ num) {
  const long long tid = (long long)blockIdx.x * blockDim.x + threadIdx.x;
  if (tid >= (long long)N_EDGES * 16) return;
  const int e = (int)(tid >> 4);
  const int f4 = (int)(tid & 15) * 4;
  const int s = src[e], d = dst[e];
  const float t = T[layer];
  const v4f ev =
      __builtin_nontemporal_load((const v4f*)(E_ + (long long)e * HDIM + f4));
  const v4f hv = *(const v4f*)(H + (long long)s * HDIM + f4);
  const long long didx = (long long)d * HDIM + f4;
  const v4i mk = *(const v4i*)(mkey + didx);
#pragma unroll
  for (int j = 0; j < 4; ++j) {
    const float msg = fmaxf(hv[j] + ev[j], 0.0f) + EPS_MSG;
    const float ex = __expf(msg * t - f32unkey(mk[j]));
    unsafeAtomicAdd(&denom[didx + j], ex);
    unsafeAtomicAdd(&num[didx + j], msg * ex);
  }
}

// h2 = h + num / (denom + 1e-16)
__global__ void agg_kernel(const float* __restrict__ H,
                           const float* __restrict__ num,
                           const float* __restrict__ denom,
                           float* __restrict__ H2, long long n) {
  const long long i = (long long)blockIdx.x * blockDim.x + threadIdx.x;
  if (i < n) H2[i] = H[i] + num[i] / (denom[i] + 1e-16f);
}

// ---------------------------------------------------------------------------
// Conv GEMM1: Z = H2 @ W1 + b1  (K=64 -> 128 cols) + per-column Σz, Σz^2.
// ---------------------------------------------------------------------------
__global__ void conv_gemm1_kernel(const float* __restrict__ H2,
                                  const float* __restrict__ W1,
                                  const float* __restrict__ B1,
                                  float* __restrict__ Z,
                                  float* __restrict__ bnsum,
                                  float* __restrict__ bnsq, int M) {
  const int wave = threadIdx.x >> 5;
  const int lane = threadIdx.x & 31;
  const int ln = lane & 15;
  const int hi = lane >> 4;
  const int kb = hi * 2;
  const int tile = blockIdx.x * 4 + wave;
  const int row0 = tile * 16;
  if (row0 >= M) return;
  const int arow = (row0 + ln < M) ? (row0 + ln) : (M - 1);

  v2f a[16];
#pragma unroll
  for (int ks = 0; ks < 16; ++ks) {
    const int kk = ks * 4 + kb;
    a[ks].x = H2[(long long)arow * HDIM + kk];
    a[ks].y = H2[(long long)arow * HDIM + kk + 1];
  }
#pragma unroll
  for (int nt = 0; nt < 8; ++nt) {
    const int col = nt * 16 + ln;
    v8f acc = splat8(B1[col]);
#pragma unroll
    for (int ks = 0; ks < 16; ++ks) {
      const int kk = ks * 4 + kb;
      v2f b;
      b.x = W1[kk * H2DIM + col];
      b.y = W1[(kk + 1) * H2DIM + col];
      acc = wmma4(a[ks], b, acc);
    }
    float s1 = 0.0f, s2 = 0.0f;
#pragma unroll
    for (int r = 0; r < 8; ++r) {
      const int grow = row0 + r + hi * 8;
      if (grow < M) {
        const float v = acc[r];
        Z[(long long)grow * H2DIM + col] = v;
        s1 += v;
        s2 += v * v;
      }
    }
    s1 += __shfl_xor(s1, 16);
    s2 += __shfl_xor(s2, 16);
    if (hi == 0) {
      unsafeAtomicAdd(&bnsum[col], s1);
      unsafeAtomicAdd(&bnsq[col], s2);
    }
  }
}

// scale/shift from batch stats
__global__ void bn_kernel(const float* __restrict__ bnsum,
                          const float* __restrict__ bnsq,
                          const float* __restrict__ gamma,
                          const float* __restrict__ beta,
                          float* __restrict__ scale, float* __restrict__ shift,
                          float invN) {
  const int j = threadIdx.x;  // 128
  const float mu = bnsum[j] * invN;
  const float var = bnsq[j] * invN - mu * mu;
  const float inv = rsqrtf(var + BN_EPS);
  const float sc = gamma[j] * inv;
  scale[j] = sc;
  shift[j] = beta[j] - mu * sc;
}

// ---------------------------------------------------------------------------
// Conv GEMM2: HOUT = [relu]( relu(Z*scale+shift) @ W2 + b2 )  (K=128 -> 64)
// ---------------------------------------------------------------------------
__global__ void conv_gemm2_kernel(const float* __restrict__ Z,
                                  const float* __restrict__ scale,
                                  const float* __restrict__ shift,
                                  const float* __restrict__ W2,
                                  const float* __restrict__ B2,
                                  float* __restrict__ HOUT, int M,
                                  int applyRelu) {
  const int wave = threadIdx.x >> 5;
  const int lane = threadIdx.x & 31;
  const int ln = lane & 15;
  const int hi = lane >> 4;
  const int kb = hi * 2;
  const int tile = blockIdx.x * 4 + wave;
  const int row0 = tile * 16;
  if (row0 >= M) return;
  const int arow = (row0 + ln < M) ? (row0 + ln) : (M - 1);

  v2f a[32];
#pragma unroll
  for (int ks = 0; ks < 32; ++ks) {
    const int kk = ks * 4 + kb;
    const float z0 = Z[(long long)arow * H2DIM + kk];
    const float z1 = Z[(long long)arow * H2DIM + kk + 1];
    a[ks].x = fmaxf(z0 * scale[kk] + shift[kk], 0.0f);
    a[ks].y = fmaxf(z1 * scale[kk + 1] + shift[kk + 1], 0.0f);
  }
#pragma unroll
  for (int nt = 0; nt < 4; ++nt) {
    const int col = nt * 16 + ln;
    v8f acc = splat8(B2[col]);
#pragma unroll
    for (int ks = 0; ks < 32; ++ks) {
      const int kk = ks * 4 + kb;
      v2f b;
      b.x = W2[kk * HDIM + col];
      b.y = W2[(kk + 1) * HDIM + col];
      acc = wmma4(a[ks], b, acc);
    }
#pragma unroll
    for (int r = 0; r < 8; ++r) {
      const int grow = row0 + r + hi * 8;
      float v = acc[r];
      if (applyRelu) v = fmaxf(v, 0.0f);
      if (grow < M) HOUT[(long long)grow * HDIM + col] = v;
    }
  }
}

// ---------------------------------------------------------------------------
// Pool + head
// ---------------------------------------------------------------------------
__global__ void pool_kernel(const float* __restrict__ H,
                            const int* __restrict__ batch,
                            float* __restrict__ G) {
  const long long tid = (long long)blockIdx.x * blockDim.x + threadIdx.x;
  if (tid >= (long long)N_NODES * HDIM) return;
  const int i = (int)(tid >> 6);
  const int f = (int)(tid & 63);
  unsafeAtomicAdd(&G[(long long)batch[i] * HDIM + f], H[tid]);
}

__global__ void head_kernel(const float* __restrict__ G,
                            const float* __restrict__ W1,
                            const float* __restrict__ B1,
                            const float* __restrict__ W2,
                            const float* __restrict__ B2,
                            float* __restrict__ OUT) {
  __shared__ float row[64];
  __shared__ float red[64];
  const int g = blockIdx.x;
  const int j = threadIdx.x;  // 64
  row[j] = G[g * 64 + j];
  __syncthreads();
  float acc = B1[j];
#pragma unroll 8
  for (int k = 0; k < 64; ++k) acc += row[k] * W1[k * 64 + j];
  red[j] = fmaxf(acc, 0.0f) * W2[j];
  __syncthreads();
  for (int s = 32; s > 0; s >>= 1) {
    if (j < s) red[j] += red[j + s];
    __syncthreads();
  }
  if (j == 0) OUT[g] = red[0] + B2[0];
}

__global__ void fill_f32(float* p, float v, long long n) {
  const long long i = (long long)blockIdx.x * blockDim.x + threadIdx.x;
  if (i < n) p[i] = v;
}
__global__ void fill_i32(int* p, int v, long long n) {
  const long long i = (long long)blockIdx.x * blockDim.x + threadIdx.x;
  if (i < n) p[i] = v;
}

// ---------------------------------------------------------------------------
extern "C" void kernel_launch(void* const* d_in, const int* in_sizes, int n_in,
                              void* d_out, int out_size, void* d_ws,
                              size_t ws_size, hipStream_t stream) {
  const float* x = (const float*)d_in[0];
  const float* edge_attr = (const float*)d_in[1];
  const int* edge_index = (const int*)d_in[2];
  const int* batch = (const int*)d_in[3];
  const float* atom_W1 = (const float*)d_in[4];
  const float* atom_b1 = (const float*)d_in[5];
  const float* atom_W2 = (const float*)d_in[6];
  const float* atom_b2 = (const float*)d_in[7];
  const float* bond_W1 = (const float*)d_in[8];
  const float* bond_b1 = (const float*)d_in[9];
  const float* bond_W2 = (const float*)d_in[10];
  const float* bond_b2 = (const float*)d_in[11];
  const float* conv_t = (const float*)d_in[12];
  const float* conv_W1 = (const float*)d_in[13];
  const float* conv_b1 = (const float*)d_in[14];
  const float* conv_gamma = (const float*)d_in[15];
  const float* conv_beta = (const float*)d_in[16];
  const float* conv_W2 = (const float*)d_in[17];
  const float* conv_b2 = (const float*)d_in[18];
  const float* lin1_W = (const float*)d_in[19];
  const float* lin1_b = (const float*)d_in[20];
  const float* lin2_W = (const float*)d_in[21];
  const float* lin2_b = (const float*)d_in[22];

  const int* src = edge_index;
  const int* dst = edge_index + N_EDGES;

  // Workspace carve-up (256B aligned)
  char* ws = (char*)d_ws;
  size_t off = 0;
  auto carve = [&](size_t bytes) -> void* {
    void* p = ws + off;
    off += (bytes + 255) & ~(size_t)255;
    return p;
  };
  float* e = (float*)carve((size_t)N_EDGES * HDIM * 4);    // 256 MB
  float* hA = (float*)carve((size_t)N_NODES * HDIM * 4);   // 12.8 MB
  float* hB = (float*)carve((size_t)N_NODES * HDIM * 4);
  float* h2 = (float*)carve((size_t)N_NODES * HDIM * 4);
  float* z = (float*)carve((size_t)N_NODES * H2DIM * 4);   // 25.6 MB
  int* mkey = (int*)carve((size_t)N_NODES * HDIM * 4);
  float* denom = (float*)carve((size_t)N_NODES * HDIM * 4);
  float* num = (float*)carve((size_t)N_NODES * HDIM * 4);
  float* bnstats = (float*)carve(256 * 4);  // [0:128)=sum, [128:256)=sumsq
  float* scale = (float*)carve(128 * 4);
  float* shift = (float*)carve(128 * 4);
  float* g = (float*)carve((size_t)NGRAPHS * HDIM * 4);
  (void)ws_size;

  const int tpb = 256;
  auto nb = [&](long long n) { return (int)((n + tpb - 1) / tpb); };
  const long long NE16 = (long long)N_EDGES * 16;  // 4 features per thread
  const long long NN64 = (long long)N_NODES * HDIM;
  const int nodeTiles = (N_NODES + 15) / 16;
  const int edgeTiles = (N_EDGES + 15) / 16;

  // Encoders (WMMA). Bond encoder streams its 256MB output non-temporally.
  encoder_kernel<8, false><<<(nodeTiles + 3) / 4, 128, 0, stream>>>(
      x, N_NODES, 32, atom_W1, atom_b1, atom_W2, atom_b2, hA);
  encoder_kernel<4, true><<<(edgeTiles + 3) / 4, 128, 0, stream>>>(
      edge_attr, N_EDGES, 16, bond_W1, bond_b1, bond_W2, bond_b2, e);

  float* hin = hA;
  float* hout = hB;
  for (int layer = 0; layer < 4; ++layer) {
    fill_i32<<<nb(NN64), tpb, 0, stream>>>(mkey, INT_MIN, NN64);
    fill_f32<<<nb(NN64), tpb, 0, stream>>>(denom, 0.0f, NN64);
    fill_f32<<<nb(NN64), tpb, 0, stream>>>(num, 0.0f, NN64);
    fill_f32<<<1, 256, 0, stream>>>(bnstats, 0.0f, 256);

    msg_max_kernel<<<nb(NE16), tpb, 0, stream>>>(hin, e, src, dst, conv_t,
                                                 layer, mkey);
    msg_sum_kernel<<<nb(NE16), tpb, 0, stream>>>(hin, e, src, dst, conv_t,
                                                 layer, mkey, denom, num);
    agg_kernel<<<nb(NN64), tpb, 0, stream>>>(hin, num, denom, h2, NN64);

    conv_gemm1_kernel<<<(nodeTiles + 3) / 4, 128, 0, stream>>>(
        h2, conv_W1 + (size_t)layer * HDIM * H2DIM, conv_b1 + layer * H2DIM, z,
        bnstats, bnstats + 128, N_NODES);
    bn_kernel<<<1, 128, 0, stream>>>(bnstats, bnstats + 128,
                                     conv_gamma + layer * H2DIM,
                                     conv_beta + layer * H2DIM, scale, shift,
                                     1.0f / (float)N_NODES);
    conv_gemm2_kernel<<<(nodeTiles + 3) / 4, 128, 0, stream>>>(
        z, scale, shift, conv_W2 + (size_t)layer * H2DIM * HDIM,
        conv_b2 + layer * HDIM, hout, N_NODES, (layer < 3) ? 1 : 0);

    float* tmp = hin;
    hin = hout;
    hout = tmp;
  }

  // Pool + head
  fill_f32<<<nb((long long)NGRAPHS * HDIM), tpb, 0, stream>>>(
      g, 0.0f, (long long)NGRAPHS * HDIM);
  pool_kernel<<<nb(NN64), tpb, 0, stream>>>(hin, batch, g);
  head_kernel<<<NGRAPHS, 64, 0, stream>>>(g, lin1_W, lin1_b, lin2_W, lin2_b,
                                          (float*)d_out);
}